// FPSWE_pool_78030965834372
// MI455X (gfx1250) — compile-verified
//
#include <hip/hip_runtime.h>
#include <hip/hip_bf16.h>

typedef __attribute__((ext_vector_type(2))) float v2f;
typedef __attribute__((ext_vector_type(8))) float v8f;

#define E_EDGES 2048
#define DEG     32
#define D_IN    128
#define N_PROJ  256
#define M_ANCH  128
#define STILE_STRIDE 258   // 8*258 % 64 == 16 -> the two lane-halves of a D store hit disjoint banks

// ---------------------------------------------------------------------------
// prep: W row-normalization into WMMA-B-friendly paired layout, plus the
// analytic interpolation-coefficient matrix BcoefT[r][p] and constant c[p].
// ---------------------------------------------------------------------------
__global__ __launch_bounds__(256) void fpswe_prep(
    const float* __restrict__ theta_v,   // [256,128]
    const float* __restrict__ weight,    // [256,128]
    const float* __restrict__ anchors,   // [128,256]
    float* __restrict__ Wpair,           // [64][256][2]  (k-pair, proj, {k,k+1})
    float* __restrict__ BcoefT,          // [32][256]
    float* __restrict__ cvec)            // [256]
{
    const int p = threadIdx.x;   // projection index 0..255

    // W = v / ||v||_row  (weight_norm with g == 1)
    float ss = 0.0f;
    #pragma unroll 8
    for (int d = 0; d < D_IN; ++d) {
        float t = theta_v[p * D_IN + d];
        ss += t * t;
    }
    const float rn = 1.0f / sqrtf(ss);
    #pragma unroll 4
    for (int kp = 0; kp < D_IN / 2; ++kp) {
        Wpair[kp * (2 * N_PROJ) + 2 * p + 0] = theta_v[p * D_IN + 2 * kp + 0] * rn;
        Wpair[kp * (2 * N_PROJ) + 2 * p + 1] = theta_v[p * D_IN + 2 * kp + 1] * rn;
    }

    // c[p] = (1/M) * sum_m weight[p,m] * anchors[m,p]
    float c = 0.0f;
    #pragma unroll 8
    for (int m = 0; m < M_ANCH; ++m)
        c += weight[p * M_ANCH + m] * anchors[m * N_PROJ + p];
    cvec[p] = c * (1.0f / (float)M_ANCH);

    // Bcoef[p,r] = sum_m w[p,m] * ((1-frac_m)*(r0_m==r) + frac_m*(r0_m==r-1))
    // with a_m = (m/(M-1))*0.99998 * (DEG-1)/0.99999, r0 = clip(floor(a),0,30)
    for (int r = 0; r < DEG; ++r) {
        float acc = 0.0f;
        for (int m = 0; m < M_ANCH; ++m) {
            float u  = ((float)m / (float)(M_ANCH - 1)) * 0.99998f;
            float a  = u * (float)(DEG - 1) / 0.99999f;
            float r0 = floorf(a);
            r0 = fminf(fmaxf(r0, 0.0f), (float)(DEG - 2));
            float fr = a - r0;
            int   ir = (int)r0;
            float w  = weight[p * M_ANCH + m];
            if (ir == r)          acc += w * (1.0f - fr);
            else if (ir == r - 1) acc += w * fr;
        }
        BcoefT[r * N_PROJ + p] = acc;
    }
}

// ---------------------------------------------------------------------------
// main fused kernel: one block per hyperedge.
//   phase 1: Xs tile (32 x 256) = Xrows (32 x 128) @ W^T via v_wmma_f32_16x16x4
//   phase 2: per-column 32-element bitonic sort + weighted reduction
// ---------------------------------------------------------------------------
__global__ __launch_bounds__(256) void fpswe_main(
    const float* __restrict__ X,        // [65536,128]
    const float* __restrict__ Wpair,    // [64][256][2]
    const float* __restrict__ BcoefT,   // [32][256]
    const float* __restrict__ cvec,     // [256]
    float* __restrict__ out)            // [2048,256]
{
    __shared__ float Stile[DEG * STILE_STRIDE];   // 33,024 B

    const int e    = blockIdx.x;
    const int lane = threadIdx.x & 31;
    const int wave = threadIdx.x >> 5;
    const int lo   = lane & 15;       // M/N index within fragment
    const int hi   = lane >> 4;       // selects K pair +2 within fragment

    const float* Xe = X + (size_t)e * DEG * D_IN;   // 32 rows of this edge

    // ----- Phase 1: WMMA GEMM -----------------------------------------------
    // wave w handles N-tiles {2w, 2w+1}, each over both M-tiles (rows 0-15, 16-31)
    for (int ntz = 0; ntz < 2; ++ntz) {
        const int nt = wave * 2 + ntz;
        const int n  = nt * 16 + lo;

        v8f acc0 = {};   // rows 0..15
        v8f acc1 = {};   // rows 16..31

        #pragma unroll 8
        for (int k2 = 0; k2 < D_IN / 4; ++k2) {        // k0 = 4*k2
            const int kk = 4 * k2 + 2 * hi;
            const int kp = 2 * k2 + hi;
            // B fragment: lane -> N, VGPR pair -> (kk, kk+1); coalesced b64 load
            v2f b = *(const v2f*)(Wpair + (size_t)kp * (2 * N_PROJ) + 2 * n);
            // A fragments: lane -> M, VGPR pair -> (kk, kk+1)
            v2f a0 = *(const v2f*)(Xe + (size_t)(lo)      * D_IN + kk);
            v2f a1 = *(const v2f*)(Xe + (size_t)(16 + lo) * D_IN + kk);
            acc0 = __builtin_amdgcn_wmma_f32_16x16x4_f32(
                false, a0, false, b, (short)0, acc0, false, false);
            acc1 = __builtin_amdgcn_wmma_f32_16x16x4_f32(
                false, a1, false, b, (short)0, acc1, false, false);
        }

        // D layout: VGPR v -> M = v + 8*hi, N = lo
        #pragma unroll
        for (int v = 0; v < 8; ++v) {
            const int m0 = v + 8 * hi;
            Stile[(m0)      * STILE_STRIDE + nt * 16 + lo] = acc0[v];
            Stile[(16 + m0) * STILE_STRIDE + nt * 16 + lo] = acc1[v];
        }
    }

    __syncthreads();

    // ----- Phase 2: per-projection sort + reduction -------------------------
    const int p = threadIdx.x;    // projection 0..255

    float s[DEG];
    #pragma unroll
    for (int r = 0; r < DEG; ++r)
        s[r] = Stile[r * STILE_STRIDE + p];   // lanes hit distinct banks (p varies)

    // fully-unrolled ascending bitonic sorting network (constant reg indices)
    #pragma unroll
    for (int k = 2; k <= DEG; k <<= 1) {
        #pragma unroll
        for (int j = k >> 1; j > 0; j >>= 1) {
            #pragma unroll
            for (int i = 0; i < DEG; ++i) {
                const int l = i ^ j;
                if (l > i) {
                    const bool up = ((i & k) == 0);
                    float a  = s[i], b = s[l];
                    float mn = fminf(a, b), mx = fmaxf(a, b);
                    s[i] = up ? mn : mx;
                    s[l] = up ? mx : mn;
                }
            }
        }
    }

    float dot = 0.0f;
    #pragma unroll
    for (int r = 0; r < DEG; ++r)
        dot += BcoefT[r * N_PROJ + p] * s[r];   // coalesced across lanes

    out[(size_t)e * N_PROJ + p] = cvec[p] - dot * (1.0f / (float)M_ANCH);
}

// ---------------------------------------------------------------------------
extern "C" void kernel_launch(void* const* d_in, const int* in_sizes, int n_in,
                              void* d_out, int out_size, void* d_ws, size_t ws_size,
                              hipStream_t stream) {
    (void)in_sizes; (void)n_in; (void)out_size; (void)ws_size;

    const float* X       = (const float*)d_in[0];
    // d_in[1] = hyperedge_index (int32) — row1 is repeat(arange(E),DEG): not needed
    const float* theta_v = (const float*)d_in[2];
    const float* weight  = (const float*)d_in[3];
    const float* anchors = (const float*)d_in[4];
    // d_in[5] = num_edges scalar (== E_EDGES, fixed by problem)

    float* ws     = (float*)d_ws;
    float* Wpair  = ws;                          // 64*512   = 32768 floats
    float* BcoefT = ws + 64 * 2 * N_PROJ;        // 32*256   =  8192 floats
    float* cvec   = BcoefT + DEG * N_PROJ;       // 256 floats

    fpswe_prep<<<1, N_PROJ, 0, stream>>>(theta_v, weight, anchors, Wpair, BcoefT, cvec);
    fpswe_main<<<E_EDGES, 256, 0, stream>>>(X, Wpair, BcoefT, cvec, (float*)d_out);
}